// TreeMPNN_57715770523748
// MI455X (gfx1250) — compile-verified
//
#include <hip/hip_runtime.h>
#include <hip/hip_bf16.h>

// ---------------------------------------------------------------------------
// TreeMPNN on gfx1250: fp32 WMMA GEMMs, async global->LDS tile staging,
// wave-per-edge GATv2 softmax/scatter.
// ---------------------------------------------------------------------------

#define N_NODES 300000
#define E_EDGES (N_NODES - 1)
#define A_ROWS  150000
#define D_IN    64
#define H_DIM   128

typedef __attribute__((ext_vector_type(2))) float v2f;
typedef __attribute__((ext_vector_type(8))) float v8f;

// float atomic-max via monotone int/uint atomics (correct for all non-NaN floats)
__device__ __forceinline__ void atomic_max_float(float* addr, float val) {
  if (val >= 0.0f) atomicMax((int*)addr, __float_as_int(val));
  else             atomicMin((unsigned int*)addr, __float_as_uint(val));
}

// ---------------------------------------------------------------------------
// Generic 16-row-stripe WMMA GEMM: C[M,128] = act(Atile[M,K] @ W[K,128] + bias)
// MODE 0: A rows read directly from Aptr [M,K]  (async DMA global->LDS)
// MODE 1: K=256, A row = concat(h[child[row]], parent<0 ? 0 : h[parent[row]])
// MODE 2: K=384, A row = concat(branch[row], time_table[tidx[row]], ctx)
// One 256-thread block = 8 waves; wave w computes output tile cols 16w..16w+15,
// rows 16*blockIdx.x .. +15.  M must be a multiple of 16 (300000, 150000 both are).
// ---------------------------------------------------------------------------
template <int K, int MODE, int RELU>
__global__ __launch_bounds__(256) void gemm_wmma(
    const float* __restrict__ Aptr, const float* __restrict__ W,
    const float* __restrict__ bias, float* __restrict__ C,
    const float* __restrict__ h, const int* __restrict__ idx_a,
    const int* __restrict__ idx_b, const float* __restrict__ t_table,
    const float* __restrict__ ctx)
{
  __shared__ float As[16 * K];
  const int row0 = blockIdx.x * 16;
  const int tid  = threadIdx.x;

  // ---- stage A tile (16 x K) into LDS ----
  if (MODE == 0) {
    // Async DMA straight into LDS, 16B per lane per transfer (ASYNCcnt path).
    // 16*K/4 chunks is a multiple of 256 for K in {64,128} -> no divergence.
    constexpr int chunksPerRow = K / 4;
    for (int c = tid; c < 16 * chunksPerRow; c += 256) {
      const int r   = c / chunksPerRow;
      const int col = (c - r * chunksPerRow) * 4;
      unsigned lds_off = (unsigned)(size_t)(&As[r * K + col]);
      const float* g = Aptr + (size_t)(row0 + r) * K + col;
      asm volatile("global_load_async_to_lds_b128 %0, %1, off"
                   :: "v"(lds_off), "v"(g) : "memory");
    }
    asm volatile("s_wait_asynccnt 0x0" ::: "memory");
  } else if (MODE == 1) {              // child | parent concat, K == 256
    for (int i = tid; i < 16 * 256; i += 256) {
      int r = i >> 8, c = i & 255;
      int row = row0 + r;
      float v;
      if (c < 128) {
        v = h[(size_t)idx_a[row] * 128 + c];
      } else {
        int p = idx_b[row];
        v = (p < 0) ? 0.0f : h[(size_t)p * 128 + (c - 128)];
      }
      As[i] = v;
    }
  } else {                             // branch | time | ctx concat, K == 384
    for (int i = tid; i < 16 * 384; i += 256) {
      int r = i / 384, c = i - r * 384;
      int row = row0 + r;
      float v;
      if (c < 128)      v = Aptr[(size_t)row * 128 + c];
      else if (c < 256) v = t_table[(size_t)idx_a[row] * 128 + (c - 128)];
      else              v = ctx[c - 256];
      As[i] = v;
    }
  }
  __syncthreads();

  const int wave = tid >> 5;
  const int lane = tid & 31;
  const int m    = lane & 15;     // row (A) / col (B) index within tile
  const int half = lane >> 4;     // K-pair selector
  const int n0   = wave * 16;     // output column base

  const float bval = bias[n0 + m];
  v8f acc;
#pragma unroll
  for (int i = 0; i < 8; ++i) acc[i] = bval;

  const float* __restrict__ Arow = &As[m * K];
#pragma unroll 4
  for (int k = 0; k < K; k += 4) {
    const int kk = k + 2 * half;
    v2f a, b;
    a.x = Arow[kk];
    a.y = Arow[kk + 1];
    b.x = W[(size_t)kk * 128 + n0 + m];
    b.y = W[(size_t)(kk + 1) * 128 + n0 + m];
    acc = __builtin_amdgcn_wmma_f32_16x16x4_f32(false, a, false, b,
                                                (short)0, acc, false, false);
  }

#pragma unroll
  for (int i = 0; i < 8; ++i) {
    float v = acc[i];
    if (RELU) v = fmaxf(v, 0.0f);
    C[(size_t)(row0 + i + 8 * half) * 128 + n0 + m] = v;
  }
}

// ---------------------------------------------------------------------------
// GATv2 edge kernels
// ---------------------------------------------------------------------------

// wave per edge: a[e] = dot(leakyrelu(xl[src]+xr[dst]), att); segment-max into amax[dst]
__global__ __launch_bounds__(256) void edge_score_kernel(
    const float* __restrict__ xl, const float* __restrict__ xr,
    const float* __restrict__ att, const int* __restrict__ src,
    const int* __restrict__ dst, float* __restrict__ a_edge,
    float* __restrict__ amax, int nE)
{
  int e = (blockIdx.x * 256 + threadIdx.x) >> 5;
  if (e >= nE) return;
  int lane = threadIdx.x & 31;
  int s = src[e], d = dst[e];
  float4 l  = ((const float4*)(xl + (size_t)s * 128))[lane];
  float4 r  = ((const float4*)(xr + (size_t)d * 128))[lane];
  float4 aa = ((const float4*)att)[lane];
  float sum = 0.0f, v;
  v = l.x + r.x; sum += (v > 0.0f ? v : 0.2f * v) * aa.x;
  v = l.y + r.y; sum += (v > 0.0f ? v : 0.2f * v) * aa.y;
  v = l.z + r.z; sum += (v > 0.0f ? v : 0.2f * v) * aa.z;
  v = l.w + r.w; sum += (v > 0.0f ? v : 0.2f * v) * aa.w;
#pragma unroll
  for (int off = 16; off > 0; off >>= 1) sum += __shfl_down(sum, off, 32);
  if (lane == 0) {
    a_edge[e] = sum;
    atomic_max_float(&amax[d], sum);
  }
}

// thread per edge: ex = exp(a - amax[dst]); denom[dst] += ex; a_edge <- ex
__global__ __launch_bounds__(256) void edge_exp_kernel(
    float* __restrict__ a_edge, const int* __restrict__ dst,
    const float* __restrict__ amax, float* __restrict__ denom, int nE)
{
  int e = blockIdx.x * 256 + threadIdx.x;
  if (e >= nE) return;
  int d = dst[e];
  float ex = expf(a_edge[e] - amax[d]);
  a_edge[e] = ex;
  atomicAdd(&denom[d], ex);
}

// wave per edge: agg[dst] += (ex/denom[dst]) * xl[src]
__global__ __launch_bounds__(256) void edge_aggregate_kernel(
    const float* __restrict__ xl, const float* __restrict__ ex_edge,
    const int* __restrict__ src, const int* __restrict__ dst,
    const float* __restrict__ denom, float* __restrict__ agg, int nE)
{
  int e = (blockIdx.x * 256 + threadIdx.x) >> 5;
  if (e >= nE) return;
  int lane = threadIdx.x & 31;
  int s = src[e], d = dst[e];
  float alpha = ex_edge[e] / (denom[d] + 1e-16f);
  float4 l = ((const float4*)(xl + (size_t)s * 128))[lane];
  float* p = agg + (size_t)d * 128 + lane * 4;
  atomicAdd(p + 0, alpha * l.x);
  atomicAdd(p + 1, alpha * l.y);
  atomicAdd(p + 2, alpha * l.z);
  atomicAdd(p + 3, alpha * l.w);
}

// h += relu(agg + bias)   (elementwise over N*128)
__global__ __launch_bounds__(256) void update_h_kernel(
    float* __restrict__ h, const float* __restrict__ agg,
    const float* __restrict__ bias, int total)
{
  int i = blockIdx.x * 256 + threadIdx.x;
  if (i < total) h[i] += fmaxf(agg[i] + bias[i & 127], 0.0f);
}

__global__ void fill_kernel(float* __restrict__ p, float v, int n) {
  int i = blockIdx.x * 256 + threadIdx.x;
  if (i < n) p[i] = v;
}

// context = focal_seq @ W_seq + b_seq   (1x64 @ 64x128)
__global__ void context_kernel(const float* __restrict__ fs,
                               const float* __restrict__ Wseq,
                               const float* __restrict__ bseq,
                               float* __restrict__ ctx)
{
  int c = threadIdx.x;   // 128 threads
  float s = bseq[c];
#pragma unroll 8
  for (int k = 0; k < 64; ++k) s += fs[k] * Wseq[k * 128 + c];
  ctx[c] = s;
}

// wave per row: out[row] = dot(hidden[row,:128], W_a2) + b_a2
__global__ __launch_bounds__(256) void logits_kernel(
    const float* __restrict__ hidden, const float* __restrict__ Wa2,
    const float* __restrict__ ba2, float* __restrict__ out, int nA)
{
  int row = (blockIdx.x * 256 + threadIdx.x) >> 5;
  if (row >= nA) return;
  int lane = threadIdx.x & 31;
  float4 hv = ((const float4*)(hidden + (size_t)row * 128))[lane];
  float4 wv = ((const float4*)Wa2)[lane];
  float sum = hv.x * wv.x + hv.y * wv.y + hv.z * wv.z + hv.w * wv.w;
#pragma unroll
  for (int off = 16; off > 0; off >>= 1) sum += __shfl_down(sum, off, 32);
  if (lane == 0) out[row] = sum + ba2[0];
}

// ---------------------------------------------------------------------------
extern "C" void kernel_launch(void* const* d_in, const int* in_sizes, int n_in,
                              void* d_out, int out_size, void* d_ws, size_t ws_size,
                              hipStream_t stream)
{
  const float* x          = (const float*)d_in[0];
  const int*   edge_index = (const int*)  d_in[1];
  const float* focal_seq  = (const float*)d_in[2];
  const int*   child_idx  = (const int*)  d_in[3];
  const int*   parent_idx = (const int*)  d_in[4];
  const int*   time_idx   = (const int*)  d_in[5];
  const float* W_embed    = (const float*)d_in[6];
  const float* b_embed    = (const float*)d_in[7];
  const float* bu_Wl  = (const float*)d_in[8];
  const float* bu_bl  = (const float*)d_in[9];
  const float* bu_Wr  = (const float*)d_in[10];
  const float* bu_br  = (const float*)d_in[11];
  const float* bu_att = (const float*)d_in[12];
  const float* bu_bias= (const float*)d_in[13];
  const float* td_Wl  = (const float*)d_in[14];
  const float* td_bl  = (const float*)d_in[15];
  const float* td_Wr  = (const float*)d_in[16];
  const float* td_br  = (const float*)d_in[17];
  const float* td_att = (const float*)d_in[18];
  const float* td_bias= (const float*)d_in[19];
  const float* time_table = (const float*)d_in[20];
  const float* W_comb = (const float*)d_in[21];
  const float* b_comb = (const float*)d_in[22];
  const float* W_a1   = (const float*)d_in[23];
  const float* b_a1   = (const float*)d_in[24];
  const float* W_a2   = (const float*)d_in[25];
  const float* b_a2   = (const float*)d_in[26];
  const float* W_seq  = (const float*)d_in[27];
  const float* b_seq  = (const float*)d_in[28];

  // workspace layout (floats)
  float* ws     = (float*)d_ws;
  float* h      = ws;                                   // N*128
  float* xl     = h  + (size_t)N_NODES * 128;           // N*128 (also: branch)
  float* xr     = xl + (size_t)N_NODES * 128;           // N*128 (also: agg, hidden)
  float* a_edge = xr + (size_t)N_NODES * 128;           // E
  float* amax   = a_edge + E_EDGES;                     // N
  float* denom  = amax + N_NODES;                       // N
  float* ctx    = denom + N_NODES;                      // 128

  const int* row0 = edge_index;            // edge_index[0]
  const int* row1 = edge_index + E_EDGES;  // edge_index[1]

  const int gN  = N_NODES / 16;            // 18750
  const int gA  = A_ROWS  / 16;            // 9375
  const int gEw = (E_EDGES + 7) / 8;       // wave-per-edge blocks
  const int gEt = (E_EDGES + 255) / 256;   // thread-per-edge blocks
  const int gNH = (N_NODES * 128) / 256;
  const int gNn = (N_NODES + 255) / 256;

  // context vector
  context_kernel<<<1, 128, 0, stream>>>(focal_seq, W_seq, b_seq, ctx);

  // h = x @ W_embed + b_embed
  gemm_wmma<64, 0, 0><<<gN, 256, 0, stream>>>(x, W_embed, b_embed, h,
      nullptr, nullptr, nullptr, nullptr, nullptr);

  // ---------------- bottom-up GATv2 (src = row1, dst = row0) ----------------
  gemm_wmma<128, 0, 0><<<gN, 256, 0, stream>>>(h, bu_Wl, bu_bl, xl,
      nullptr, nullptr, nullptr, nullptr, nullptr);
  gemm_wmma<128, 0, 0><<<gN, 256, 0, stream>>>(h, bu_Wr, bu_br, xr,
      nullptr, nullptr, nullptr, nullptr, nullptr);
  fill_kernel<<<gNn, 256, 0, stream>>>(amax, -__builtin_huge_valf(), N_NODES);
  hipMemsetAsync(denom, 0, (size_t)N_NODES * sizeof(float), stream);
  edge_score_kernel<<<gEw, 256, 0, stream>>>(xl, xr, bu_att, row1, row0,
                                             a_edge, amax, E_EDGES);
  edge_exp_kernel<<<gEt, 256, 0, stream>>>(a_edge, row0, amax, denom, E_EDGES);
  hipMemsetAsync(xr, 0, (size_t)N_NODES * 128 * sizeof(float), stream); // xr -> agg
  edge_aggregate_kernel<<<gEw, 256, 0, stream>>>(xl, a_edge, row1, row0,
                                                 denom, xr, E_EDGES);
  update_h_kernel<<<gNH, 256, 0, stream>>>(h, xr, bu_bias, N_NODES * 128);

  // ---------------- top-down GATv2 (src = row0, dst = row1) ----------------
  gemm_wmma<128, 0, 0><<<gN, 256, 0, stream>>>(h, td_Wl, td_bl, xl,
      nullptr, nullptr, nullptr, nullptr, nullptr);
  gemm_wmma<128, 0, 0><<<gN, 256, 0, stream>>>(h, td_Wr, td_br, xr,
      nullptr, nullptr, nullptr, nullptr, nullptr);
  fill_kernel<<<gNn, 256, 0, stream>>>(amax, -__builtin_huge_valf(), N_NODES);
  hipMemsetAsync(denom, 0, (size_t)N_NODES * sizeof(float), stream);
  edge_score_kernel<<<gEw, 256, 0, stream>>>(xl, xr, td_att, row0, row1,
                                             a_edge, amax, E_EDGES);
  edge_exp_kernel<<<gEt, 256, 0, stream>>>(a_edge, row1, amax, denom, E_EDGES);
  hipMemsetAsync(xr, 0, (size_t)N_NODES * 128 * sizeof(float), stream); // xr -> agg
  edge_aggregate_kernel<<<gEw, 256, 0, stream>>>(xl, a_edge, row0, row1,
                                                 denom, xr, E_EDGES);
  update_h_kernel<<<gNH, 256, 0, stream>>>(h, xr, td_bias, N_NODES * 128);

  // ---------------- head ----------------
  // branch = [h[child] | h[parent]] @ W_comb + b_comb   -> xl
  gemm_wmma<256, 1, 0><<<gA, 256, 0, stream>>>(nullptr, W_comb, b_comb, xl,
      h, child_idx, parent_idx, nullptr, nullptr);
  // hidden = relu([branch | t_emb | ctx] @ W_a1 + b_a1) -> xr
  gemm_wmma<384, 2, 1><<<gA, 256, 0, stream>>>(xl, W_a1, b_a1, xr,
      nullptr, time_idx, nullptr, time_table, ctx);
  // logits = hidden @ W_a2 + b_a2
  logits_kernel<<<(A_ROWS + 7) / 8, 256, 0, stream>>>(xr, W_a2, b_a2,
                                                      (float*)d_out, A_ROWS);
}